// CapsuleLayer_50972671869433
// MI455X (gfx1250) — compile-verified
//
#include <hip/hip_runtime.h>
#include <math.h>

typedef __attribute__((ext_vector_type(2))) float v2f;
typedef __attribute__((ext_vector_type(8))) float v8f;

#define B_   32
#define J_   32
#define I_   4608
#define DIN  8
#define DOUT 16
#define CHUNK 16
#define NBLK (I_ / CHUNK)          // 288 blocks
#define ROUT (B_ * J_ * DOUT)      // 16384 floats

// ---------------------------------------------------------------------------
// zero a float buffer
// ---------------------------------------------------------------------------
__global__ void caps_zero(float* p, int n) {
  int k = blockIdx.x * blockDim.x + threadIdx.x;
  if (k < n) p[k] = 0.f;
}

// ---------------------------------------------------------------------------
// One fused routing round:
//   for each i in this block's chunk:
//     u_hat[:, :, i, :]  = inputs[:, i, :] @ W[:, i, :, :]^T   (WMMA f32 16x16x4)
//     logits[b, j]       = dot(u_hat[b, j, i, :], v_accum[b, j, :])
//     c[b, :]            = softmax_j(logits)
//     s[b, j, :]        += c[b, j] * u_hat[b, j, i, :]
//   atomically flush block-partial s into global s.
// v_accum == 0 on round 0  ->  logits == 0  ->  uniform softmax == 1/J  (exact).
// ---------------------------------------------------------------------------
__global__ __launch_bounds__(256)
void caps_pass(const float* __restrict__ inp, const float* __restrict__ Wt,
               const float* __restrict__ vacc, float* __restrict__ sout)
{
  extern __shared__ float U[];           // 16384 floats: u_hat tile [b][j][d]

  const int t     = threadIdx.x;
  const int wave  = t >> 5;
  const int lane  = t & 31;
  const int lhalf = lane >> 4;           // 0: lanes 0-15, 1: lanes 16-31
  const int l16   = lane & 15;

  // routing ownership: thread owns (b = rb, d in {rg, rg+8}) for all j
  const int rb = t >> 3;                 // 0..31
  const int rg = t & 7;                  // 0..7

  float va0[J_], va1[J_], s0[J_], s1[J_];
  #pragma unroll
  for (int j = 0; j < J_; ++j) {
    const int base = (rb * J_ + j) * DOUT;
    va0[j] = vacc[base + rg];
    va1[j] = vacc[base + rg + 8];
    s0[j] = 0.f;
    s1[j] = 0.f;
  }

  const int iStart = blockIdx.x * CHUNK;

  for (int ii = 0; ii < CHUNK; ++ii) {
    const int i = iStart + ii;

    // ---- WMMA phase: 64 (Mtile, j) units across 8 waves ------------------
    #pragma unroll
    for (int k = 0; k < 8; ++k) {
      const int u    = (wave << 3) | k;  // 0..63
      const int j    = u & 31;
      const int Mt   = u >> 5;           // 0 or 1
      const int brow = l16 + (Mt << 4);  // A-matrix row (batch index)

      // A: inputs[b, i, a]; lane holds (M=l16, K = 2*lhalf + {0,1}) per K-step
      const float* pa = inp + ((size_t)brow * I_ + i) * DIN + (lhalf << 1);
      // B: W[j, i, d, a] as B[K=a][N=d]; lane holds (K = 2*lhalf + {0,1}, N=l16)
      const float* pb = Wt + (((size_t)j * I_ + i) * DOUT + l16) * DIN + (lhalf << 1);

      // prefetch next i's W tile slice (i stride = DOUT*DIN floats)
      __builtin_prefetch(pb + DOUT * DIN, 0, 0);

      v2f a0 = *(const v2f*)(pa);        // K-step 0: a = {0..3}
      v2f a1 = *(const v2f*)(pa + 4);    // K-step 1: a = {4..7}
      v2f b0 = *(const v2f*)(pb);
      v2f b1 = *(const v2f*)(pb + 4);

      v8f acc = {0.f, 0.f, 0.f, 0.f, 0.f, 0.f, 0.f, 0.f};
      acc = __builtin_amdgcn_wmma_f32_16x16x4_f32(false, a0, false, b0,
                                                  (short)0, acc, false, false);
      acc = __builtin_amdgcn_wmma_f32_16x16x4_f32(false, a1, false, b1,
                                                  (short)0, acc, false, false);

      // C layout: VGPR v -> M = v + 8*lhalf (+16*Mt), N = l16
      #pragma unroll
      for (int v = 0; v < 8; ++v) {
        const int b_out = v + (lhalf << 3) + (Mt << 4);
        U[(b_out * J_ + j) * DOUT + l16] = acc[v];
      }
    }
    __syncthreads();

    // ---- routing phase ---------------------------------------------------
    float lg[J_];
    #pragma unroll
    for (int j = 0; j < J_; ++j) {
      const int base = (rb * J_ + j) * DOUT;
      lg[j] = U[base + rg] * va0[j] + U[base + rg + 8] * va1[j];
    }
    // reduce partial dots across the 8-thread group owning this b
    #pragma unroll
    for (int m = 1; m <= 4; m <<= 1) {
      #pragma unroll
      for (int j = 0; j < J_; ++j) lg[j] += __shfl_xor(lg[j], m, 8);
    }

    float mx = lg[0];
    #pragma unroll
    for (int j = 1; j < J_; ++j) mx = fmaxf(mx, lg[j]);
    float den = 0.f;
    #pragma unroll
    for (int j = 0; j < J_; ++j) den += __expf(lg[j] - mx);
    const float inv = 1.f / den;

    #pragma unroll
    for (int j = 0; j < J_; ++j) {
      const float c    = __expf(lg[j] - mx) * inv;
      const int   base = (rb * J_ + j) * DOUT;
      s0[j] += c * U[base + rg];
      s1[j] += c * U[base + rg + 8];
    }
    __syncthreads();   // U is rewritten next iteration
  }

  // ---- flush block-partial s into global accumulator ---------------------
  #pragma unroll
  for (int j = 0; j < J_; ++j) {
    const int base = (rb * J_ + j) * DOUT;
    atomicAdd(&sout[base + rg],     s0[j]);
    atomicAdd(&sout[base + rg + 8], s1[j]);
  }
}

// ---------------------------------------------------------------------------
// v = squash(s); vout = v; vacc += v; s = 0 (ready for next round)
// ---------------------------------------------------------------------------
__global__ void caps_squash(float* __restrict__ s, float* __restrict__ vacc,
                            float* __restrict__ vout)
{
  const int idx = blockIdx.x * blockDim.x + threadIdx.x;  // (b*J + j)
  if (idx >= B_ * J_) return;

  float sv[DOUT];
  float sn = 0.f;
  #pragma unroll
  for (int d = 0; d < DOUT; ++d) {
    sv[d] = s[idx * DOUT + d];
    sn += sv[d] * sv[d];
  }
  const float scale = (sn / (1.f + sn)) / sqrtf(sn + 1e-7f);
  #pragma unroll
  for (int d = 0; d < DOUT; ++d) {
    const float v = scale * sv[d];
    vout[idx * DOUT + d]  = v;
    vacc[idx * DOUT + d] += v;   // logits accumulate dot(u_hat, sum of v's)
    s[idx * DOUT + d]     = 0.f; // re-zero for next routing round
  }
}

// ---------------------------------------------------------------------------
extern "C" void kernel_launch(void* const* d_in, const int* in_sizes, int n_in,
                              void* d_out, int out_size, void* d_ws, size_t ws_size,
                              hipStream_t stream)
{
  const float* inp = (const float*)d_in[0];   // [32, 4608, 8]
  const float* Wt  = (const float*)d_in[1];   // [32, 4608, 16, 8]
  float* sbuf = (float*)d_ws;                 // 16384 floats
  float* vacc = sbuf + ROUT;                  // 16384 floats
  float* vout = (float*)d_out;                // [32, 32, 16]

  caps_zero<<<(2 * ROUT + 255) / 256, 256, 0, stream>>>(sbuf, 2 * ROUT);

  for (int r = 0; r < 3; ++r) {
    caps_pass<<<NBLK, 256, ROUT * sizeof(float), stream>>>(inp, Wt, vacc, sbuf);
    caps_squash<<<4, 256, 0, stream>>>(sbuf, vacc, vout);
  }
}